// GConv_15118284882190
// MI455X (gfx1250) — compile-verified
//
#include <hip/hip_runtime.h>

#define NN 100000
#define NE 1600000
#define DD 128
#define LL 3
#define NG 512
#define BN_EPS 1e-5f

typedef __attribute__((ext_vector_type(2))) float v2f;
typedef __attribute__((ext_vector_type(8))) float v8f;

// ---------------- zero ----------------
__global__ void zero_f32(float* __restrict__ p, long n) {
    long i = (long)blockIdx.x * blockDim.x + threadIdx.x;
    if (i < n) p[i] = 0.f;
}

// ---------------- agg = z (strided copy) ----------------
__global__ void copy_strided(const float* __restrict__ z, int zstride,
                             float* __restrict__ agg) {
    long i = (long)blockIdx.x * blockDim.x + threadIdx.x;   // over NN*DD
    if (i >= (long)NN * DD) return;
    long r = i >> 7;
    int  c = (int)(i & 127);
    agg[i] = z[r * (size_t)zstride + c];
}

// ---------------- edge scatter-add: agg[dst] += z[src] ----------------
// one wave per edge; each lane moves 4 consecutive floats (float4 gather)
__global__ __launch_bounds__(256) void edge_scatter(
    const float* __restrict__ z, int zstride,
    const int* __restrict__ src, const int* __restrict__ dst,
    float* __restrict__ agg) {
    long gtid = (long)blockIdx.x * blockDim.x + threadIdx.x;
    long e = gtid >> 5;
    int lane = (int)(gtid & 31);
    if (e >= NE) return;
    int s = src[e], d = dst[e];
    const float4 v = *reinterpret_cast<const float4*>(z + (size_t)s * zstride + lane * 4);
    float* out = agg + (size_t)d * DD + lane * 4;
    atomicAdd(out + 0, v.x);
    atomicAdd(out + 1, v.y);
    atomicAdd(out + 2, v.z);
    atomicAdd(out + 3, v.w);
}

// ---------------- C = act(A @ W + bias), A:[NN,128] W:[128,128] ----------------
// grid.x = NN/16 = 6250 blocks; 256 threads = 8 wave32; wave w owns N-tile w*16.
// K-loop: 32x v_wmma_f32_16x16x4_f32.
__global__ __launch_bounds__(256) void gemm_bias_act(
    const float* __restrict__ A,
    const float* __restrict__ W,
    const float* __restrict__ bias,
    float* __restrict__ C, int relu) {
    __shared__ float As[16 * 132];              // padded stride to spread LDS banks
    const int lane  = threadIdx.x & 31;
    const int wave  = threadIdx.x >> 5;         // 0..7
    const int mbase = blockIdx.x * 16;
    const int nbase = wave * 16;

    // cooperative stage of 16x128 A tile into LDS (float2 granularity)
    for (int t = threadIdx.x; t < 16 * 64; t += 256) {
        int r  = t >> 6;          // 64 float2 per row
        int c2 = t & 63;
        const float2 v = *reinterpret_cast<const float2*>(
            A + (size_t)(mbase + r) * DD + c2 * 2);
        *reinterpret_cast<float2*>(&As[r * 132 + c2 * 2]) = v;
    }
    __syncthreads();

    const int l15   = lane & 15;      // A row within tile / C col within tile
    const int khalf = (lane >> 4) * 2;

    v8f acc = {};
    #pragma unroll
    for (int k = 0; k < 128; k += 4) {
        const int kk = k + khalf;
        v2f a;                                   // A frag: 16x4 f32 layout
        a.x = As[l15 * 132 + kk];
        a.y = As[l15 * 132 + kk + 1];
        v2f b;                                   // B frag: 4x16 f32 layout
        b.x = W[(size_t)kk * DD + nbase + l15];
        b.y = W[(size_t)(kk + 1) * DD + nbase + l15];
        acc = __builtin_amdgcn_wmma_f32_16x16x4_f32(
            false, a, false, b, (short)0, acc, false, false);
    }

    const float bv = bias[nbase + l15];
    const int rowoff = (lane >> 4) * 8;          // C: lanes 16-31 hold M=r+8
    #pragma unroll
    for (int r = 0; r < 8; ++r) {
        float v = acc[r] + bv;
        if (relu) v = fmaxf(v, 0.f);
        C[(size_t)(mbase + r + rowoff) * DD + nbase + l15] = v;
    }
}

// ---------------- per-column sum / sumsq ----------------
__global__ __launch_bounds__(128) void col_stats(
    const float* __restrict__ h, float* __restrict__ sum,
    float* __restrict__ sumsq) {
    const int d = threadIdx.x;
    float s = 0.f, ss = 0.f;
    for (int i = blockIdx.x; i < NN; i += gridDim.x) {
        float v = h[(size_t)i * DD + d];
        s  += v;
        ss += v * v;
    }
    atomicAdd(&sum[d], s);
    atomicAdd(&sumsq[d], ss);
}

// ---------------- BN (train stats) + optional ReLU + write z_cat + pool g_cat --
__global__ __launch_bounds__(256) void bn_pool(
    const float* __restrict__ h,
    const float* __restrict__ sum, const float* __restrict__ sumsq,
    const float* __restrict__ gamma, const float* __restrict__ beta,
    const int* __restrict__ batch,
    float* __restrict__ zcat,     // d_out + l*128 (row stride L*D)
    float* __restrict__ gcat,     // d_out + NN*L*D + l*128 (row stride L*D)
    int relu) {
    long i = (long)blockIdx.x * blockDim.x + threadIdx.x;   // over NN*DD
    if (i >= (long)NN * DD) return;
    long r = i >> 7;
    int  d = (int)(i & 127);
    const float inv_n = 1.f / (float)NN;
    float mean = sum[d] * inv_n;
    float var  = sumsq[d] * inv_n - mean * mean;
    float v = gamma[d] * (h[i] - mean) * rsqrtf(var + BN_EPS) + beta[d];
    if (relu) v = fmaxf(v, 0.f);
    zcat[r * (size_t)(LL * DD) + d] = v;
    atomicAdd(&gcat[(size_t)batch[r] * (LL * DD) + d], v);
}

extern "C" void kernel_launch(void* const* d_in, const int* in_sizes, int n_in,
                              void* d_out, int out_size, void* d_ws, size_t ws_size,
                              hipStream_t stream) {
    const float* x     = (const float*)d_in[0];
    const int*   ei    = (const int*)d_in[1];
    const int*   src   = ei;
    const int*   dst   = ei + NE;
    const int*   batch = (const int*)d_in[2];
    const float* W1    = (const float*)d_in[3];
    const float* b1    = (const float*)d_in[4];
    const float* W2    = (const float*)d_in[5];
    const float* b2    = (const float*)d_in[6];
    const float* gamma = (const float*)d_in[7];
    const float* beta  = (const float*)d_in[8];

    float* out  = (float*)d_out;
    float* zcat = out;                                  // [NN, L*D]
    float* gcat = out + (size_t)NN * (LL * DD);         // [NG, L*D]

    float* agg   = (float*)d_ws;                        // [NN, D] (also reused as h)
    float* h1    = agg + (size_t)NN * DD;               // [NN, D]
    float* stats = h1 + (size_t)NN * DD;                // 256 floats
    float* csum  = stats;
    float* csq   = stats + DD;

    const long ND = (long)NN * DD;
    const int  TB = 256;

    // zero the graph-pool output region (accumulated via atomics)
    {
        long n = (long)NG * (LL * DD);
        zero_f32<<<(int)((n + TB - 1) / TB), TB, 0, stream>>>(gcat, n);
    }

    for (int l = 0; l < LL; ++l) {
        const float* z  = (l == 0) ? x : (zcat + (size_t)(l - 1) * DD);
        const int    zs = (l == 0) ? DD : (LL * DD);

        // agg = z
        copy_strided<<<(int)((ND + TB - 1) / TB), TB, 0, stream>>>(z, zs, agg);
        // agg[dst] += z[src] over all edges (one wave per edge)
        {
            long threads = (long)NE * 32;
            edge_scatter<<<(int)((threads + TB - 1) / TB), TB, 0, stream>>>(
                z, zs, src, dst, agg);
        }
        // h1 = relu(agg @ W1[l] + b1[l])
        gemm_bias_act<<<NN / 16, TB, 0, stream>>>(
            agg, W1 + (size_t)l * DD * DD, b1 + (size_t)l * DD, h1, 1);
        // h = h1 @ W2[l] + b2[l]   (stored back into agg buffer)
        gemm_bias_act<<<NN / 16, TB, 0, stream>>>(
            h1, W2 + (size_t)l * DD * DD, b2 + (size_t)l * DD, agg, 0);
        // batch-norm statistics
        zero_f32<<<1, 256, 0, stream>>>(stats, 256);
        col_stats<<<512, 128, 0, stream>>>(agg, csum, csq);
        // normalize (+ReLU on l<L-1), write z_cat slice, pool into g_cat slice
        bn_pool<<<(int)((ND + TB - 1) / TB), TB, 0, stream>>>(
            agg, csum, csq, gamma + (size_t)l * DD, beta + (size_t)l * DD,
            batch, zcat + (size_t)l * DD, gcat + (size_t)l * DD,
            (l < LL - 1) ? 1 : 0);
    }
}